// AttentionAugmentation2D_33225867001990
// MI455X (gfx1250) — compile-verified
//
#include <hip/hip_runtime.h>
#include <hip/hip_bf16.h>

typedef float v2f __attribute__((ext_vector_type(2)));
typedef float v8f __attribute__((ext_vector_type(8)));

#define BATCH 8
#define IMH 32
#define IMW 32
#define CIN 256
#define NHEAD 8
#define HW 1024
#define TKEY 64   // keys staged per TDM transfer

#if defined(__has_builtin)
#  if __has_builtin(__builtin_amdgcn_tensor_load_to_lds)
#    define HAVE_TDM 1
#  endif
#endif
#ifndef HAVE_TDM
#  define HAVE_TDM 0
#endif

// ---------------------------------------------------------------------------
// Depthwise 3x3 conv (SAME padding), NHWC, per-channel kernel (3,3,C)
// ---------------------------------------------------------------------------
__global__ __launch_bounds__(256) void dw_conv3x3(const float* __restrict__ x,
                                                  const float* __restrict__ k,
                                                  float* __restrict__ out) {
  int gid = blockIdx.x * 256 + threadIdx.x;   // over B*H*W*C
  int c  = gid & (CIN - 1);
  int t  = gid >> 8;
  int xw = t & (IMW - 1);
  int y  = (t >> 5) & (IMH - 1);
  int b  = t >> 10;
  float acc = 0.f;
#pragma unroll
  for (int dy = -1; dy <= 1; ++dy) {
    int yy = y + dy;
    if (yy < 0 || yy >= IMH) continue;
#pragma unroll
    for (int dx = -1; dx <= 1; ++dx) {
      int xx = xw + dx;
      if (xx < 0 || xx >= IMW) continue;
      acc += x[((size_t)((b * IMH + yy) * IMW + xx) * CIN) + c] *
             k[((dy + 1) * 3 + (dx + 1)) * CIN + c];
    }
  }
  out[gid] = acc;
}

// ---------------------------------------------------------------------------
// WMMA fp32 GEMM + bias:  C[m, col0+n0+n] = sum_k A[m,k] * W[k,n0+n] + bias[n0+n]
// Grid: (N/16, M/128). Block 256 = 8 waves; wave w owns rows m0+16w .. +15.
// ---------------------------------------------------------------------------
__global__ __launch_bounds__(256) void gemm_bias_wmma(const float* __restrict__ A, int lda,
                                                      const float* __restrict__ Wt, int N,
                                                      const float* __restrict__ bias,
                                                      float* __restrict__ C, int ldc,
                                                      int col0, int K) {
  __shared__ float Wbuf[256 * 16];            // column panel, [k][n] stride 16
  const int tid = threadIdx.x;
  const int n0 = blockIdx.x * 16;
  const int total = K * 16;
  for (int idx = tid; idx < total; idx += 256)
    Wbuf[idx] = Wt[(size_t)(idx >> 4) * N + n0 + (idx & 15)];
  __syncthreads();

  const int wv = tid >> 5;
  const int lane = tid & 31;
  const int half = lane >> 4;                 // operand K-half select
  const int ln = lane & 15;                   // A row / B col within tile
  const int m0 = blockIdx.y * 128 + wv * 16;
  const float* arow = A + (size_t)(m0 + ln) * lda;

  v8f acc = {0.f, 0.f, 0.f, 0.f, 0.f, 0.f, 0.f, 0.f};
  const int nch = K >> 2;
  for (int t = 0; t < nch; ++t) {
    const int kb = 4 * t + 2 * half;
    __builtin_prefetch(arow + kb + 64, 0, 1); // global_prefetch_b8
    v2f a = *(const v2f*)(arow + kb);
    v2f bb;
    bb.x = Wbuf[kb * 16 + ln];
    bb.y = Wbuf[(kb + 1) * 16 + ln];
    acc = __builtin_amdgcn_wmma_f32_16x16x4_f32(false, a, false, bb,
                                                (short)0, acc, false, false);
  }
  const float bn = bias[n0 + ln];
#pragma unroll
  for (int r = 0; r < 8; ++r) {
    const int m = m0 + r + 8 * half;
    C[(size_t)m * ldc + col0 + n0 + ln] = acc[r] + bn;
  }
}

// ---------------------------------------------------------------------------
// TDM: one 3D descriptor stages K(64x16) + V(64x16) for one key-stage:
//   x = 16 ch (contig 64B), y = 64 rows (stride 384 f32), z = 2 planes (+128 f32)
// LDS pad: +2 DWORDs every 16 DWORDs -> row stride 18 floats (bank-friendly,
// 8B aligned for v2f operand reads); V lands at float offset 64*18.
// ---------------------------------------------------------------------------
#if HAVE_TDM
typedef unsigned int v4u __attribute__((ext_vector_type(4)));
typedef int v4i __attribute__((ext_vector_type(4)));
typedef int v8i __attribute__((ext_vector_type(8)));

__device__ __forceinline__ void tdm_load_stage(const float* kvplane, unsigned lds_off,
                                               int stage) {
  unsigned long long ga =
      (unsigned long long)(kvplane + (size_t)stage * TKEY * 384);
  v4u g0;
  g0.x = 1u;                                   // count=1, user mode
  g0.y = lds_off;                              // D#.lds_addr (bytes)
  g0.z = (unsigned)ga;                         // global_addr[31:0]
  g0.w = (unsigned)((ga >> 32) & 0x01FFFFFFu) | 0x80000000u;  // [56:32] | type=2
  v8i g1;
  g1[0] = (int)((2u << 16)        // data_size = 4B
              | (1u << 20)        // pad_enable
              | (3u << 22)        // pad_interval: 16 DWORDs
              | (1u << 25));      // pad_amount: 2 DWORDs
  g1[1] = (int)(16u << 16);       // tensor_dim0 = 16 (low16 at bits 63:48)
  g1[2] = (int)((unsigned)HW << 16); // tensor_dim1 = 1024 (low16 at 95:80)
  g1[3] = (int)(16u << 16);       // tile_dim0 = 16 (bits 127:112)
  g1[4] = (int)(TKEY | (2 << 16));// tile_dim1 = 64, tile_dim2 = 2
  g1[5] = 384;                    // tensor_dim0_stride low32
  g1[6] = (int)(128u << 16);      // td0_stride hi16=0 | tensor_dim1_stride lo16=128
  g1[7] = 0;                      // tensor_dim1_stride high32
  v4i g2 = {2, 0, 0, 0};          // tensor_dim2 = 2
  v4i g3 = {0, 0, 0, 0};
#if __clang_major__ >= 23
  v8i g4 = {0, 0, 0, 0, 0, 0, 0, 0};
  __builtin_amdgcn_tensor_load_to_lds(g0, g1, g2, g3, g4, 0);
#else
  __builtin_amdgcn_tensor_load_to_lds(g0, g1, g2, g3, 0);
#endif
}
#endif

// ---------------------------------------------------------------------------
// Fused flash attention with relative logits, fp32 WMMA, TDM-staged K/V.
// Grid: (HW/128, B*NH). Block 256 = 8 waves; wave owns 16 queries.
// logits[n,m] = q_n . k_m + RW[n][m%32] + RH[n][m/32]   (q pre-scaled 1/4)
// ---------------------------------------------------------------------------
__global__ __launch_bounds__(256) void attn_flash_wmma(const float* __restrict__ qkv,
                                                       const float* __restrict__ relw,
                                                       const float* __restrict__ relh,
                                                       float* __restrict__ attnout) {
  __shared__ float KVbuf[2][2 * TKEY * 18];   // [buf][ K 64x18 | V 64x18 ]
  __shared__ float Pbuf[128 * 18];            // per-wave P staging (C->A relayout)
  __shared__ float RW[128 * 32];
  __shared__ float RH[128 * 32];

  const int tid = threadIdx.x;
  const int wv = tid >> 5;
  const int lane = tid & 31;
  const int half = lane >> 4;
  const int ln = lane & 15;
  const int bh = blockIdx.y;
  const int b = bh >> 3;
  const int h = bh & 7;
  const int q0 = blockIdx.x * 128 + wv * 16;
  const int NSTG = HW / TKEY;                 // 16 stages

  const float* kvbase = qkv + (size_t)(b * HW) * 384 + 128 + h * 16; // K plane; V at +128

#if HAVE_TDM
  if (wv == 0)                                // kick off stage 0 ASAP (overlaps RW/RH)
    tdm_load_stage(kvbase, (unsigned)(size_t)(const void*)&KVbuf[0][0], 0);
#endif

  // --- this lane's query row (A row m = ln), pre-scaled by dkh^-0.5 = 0.25 ---
  const int nq = q0 + ln;
  const float* qrow = qkv + (size_t)(b * HW + nq) * 384 + h * 16;
  float qv[16];
#pragma unroll
  for (int d = 0; d < 16; ++d) qv[d] = qrow[d] * 0.25f;

  // --- relative-logit tables (wave-private rows; same-wave DS is in-order) ---
  const int ym = nq & 31, xm = nq >> 5;
  const int lq = wv * 16 + ln;
  for (int j = 0; j < 16; ++j) {
    const int pk = half * 16 + j;
    const float* rw = relw + (pk - ym + 31) * 16;
    const float* rh = relh + (pk - xm + 31) * 16;
    float sw = 0.f, sh = 0.f;
#pragma unroll
    for (int d = 0; d < 16; ++d) { sw += qv[d] * rw[d]; sh += qv[d] * rh[d]; }
    RW[lq * 32 + pk] = sw;
    RH[lq * 32 + pk] = sh;
  }

  // --- Q A-operand: element e of chunk t holds K = 4t + 2*half + e ---
  v2f aq[4];
#pragma unroll
  for (int t = 0; t < 4; ++t) {
    aq[t].x = half ? qv[4 * t + 2] : qv[4 * t + 0];
    aq[t].y = half ? qv[4 * t + 3] : qv[4 * t + 1];
  }

  v8f o = {0.f, 0.f, 0.f, 0.f, 0.f, 0.f, 0.f, 0.f};
  float mprev[8], lsum[8];
#pragma unroll
  for (int r = 0; r < 8; ++r) { mprev[r] = -3.0e38f; lsum[r] = 0.f; }

#if HAVE_TDM
  if (wv == 0) __builtin_amdgcn_s_wait_tensorcnt(0);
  __syncthreads();                            // stage 0 resident
#endif

  for (int stg = 0; stg < NSTG; ++stg) {
    const int cur = stg & 1;
#if HAVE_TDM
    if (wv == 0 && stg + 1 < NSTG)            // async prefetch of next stage
      tdm_load_stage(kvbase,
                     (unsigned)(size_t)(const void*)&KVbuf[cur ^ 1][0], stg + 1);
    const float* Ks = &KVbuf[cur][0];
    const float* Vs = &KVbuf[cur][TKEY * 18];
#else
    __syncthreads();
    for (int i = tid; i < 2 * TKEY * 16; i += 256) {
      const int kvp = (i >= TKEY * 16) ? 1 : 0;
      const int j = i - kvp * TKEY * 16;
      const int r = j >> 4, d = j & 15;
      KVbuf[0][kvp * TKEY * 18 + r * 18 + d] =
          kvbase[(size_t)(stg * TKEY + r) * 384 + kvp * 128 + d];
    }
    __syncthreads();
    const float* Ks = &KVbuf[0][0];
    const float* Vs = &KVbuf[0][TKEY * 18];
#endif

#pragma unroll
    for (int sub = 0; sub < 4; ++sub) {
      const float* Kt = Ks + sub * 16 * 18;
      const float* Vt = Vs + sub * 16 * 18;

      // S = Q @ K^T (K rows one-per-lane == B operand of K^T)
      v8f sc = {0.f, 0.f, 0.f, 0.f, 0.f, 0.f, 0.f, 0.f};
#pragma unroll
      for (int t = 0; t < 4; ++t) {
        const int kb = 4 * t + 2 * half;
        v2f bb = *(const v2f*)(&Kt[ln * 18 + kb]);
        sc = __builtin_amdgcn_wmma_f32_16x16x4_f32(false, aq[t], false, bb,
                                                   (short)0, sc, false, false);
      }

      // rel logits + online softmax (row m = r + 8*half, col = ln)
      const int ng = stg * TKEY + sub * 16 + ln;
      const int yk = ng & 31, xk = ng >> 5;
#pragma unroll
      for (int r = 0; r < 8; ++r) {
        const int lqr = wv * 16 + r + 8 * half;
        float sv = sc[r] + RW[lqr * 32 + yk] + RH[lqr * 32 + xk];
        float mt = sv;
        mt = fmaxf(mt, __shfl_xor(mt, 1, 32));
        mt = fmaxf(mt, __shfl_xor(mt, 2, 32));
        mt = fmaxf(mt, __shfl_xor(mt, 4, 32));
        mt = fmaxf(mt, __shfl_xor(mt, 8, 32));
        const float mnew = fmaxf(mprev[r], mt);
        const float corr = __expf(mprev[r] - mnew);
        const float p = __expf(sv - mnew);
        float ps = p;
        ps += __shfl_xor(ps, 1, 32);
        ps += __shfl_xor(ps, 2, 32);
        ps += __shfl_xor(ps, 4, 32);
        ps += __shfl_xor(ps, 8, 32);
        lsum[r] = lsum[r] * corr + ps;
        mprev[r] = mnew;
        o[r] *= corr;
        Pbuf[(size_t)(wv * 16 + r + 8 * half) * 18 + ln] = p;
      }

      // O += P @ V (P read back in A layout; same-wave DS in-order)
#pragma unroll
      for (int t = 0; t < 4; ++t) {
        const int kb = 4 * t + 2 * half;
        v2f pa = *(const v2f*)(&Pbuf[(wv * 16 + ln) * 18 + kb]);
        v2f vb;
        vb.x = Vt[kb * 18 + ln];
        vb.y = Vt[(kb + 1) * 18 + ln];
        o = __builtin_amdgcn_wmma_f32_16x16x4_f32(false, pa, false, vb,
                                                  (short)0, o, false, false);
      }
    }

#if HAVE_TDM
    if (wv == 0 && stg + 1 < NSTG) __builtin_amdgcn_s_wait_tensorcnt(0);
    __syncthreads();   // publishes next stage; guards buffer reuse
#endif
  }

  // write normalized O to (B, HW, NH*DVH), channel = h*16 + d
#pragma unroll
  for (int r = 0; r < 8; ++r) {
    const int m = r + 8 * half;
    attnout[(size_t)(b * HW + q0 + m) * 128 + h * 16 + ln] = o[r] / lsum[r];
  }
}

// ---------------------------------------------------------------------------
// Host-side launcher
// ---------------------------------------------------------------------------
extern "C" void kernel_launch(void* const* d_in, const int* in_sizes, int n_in,
                              void* d_out, int out_size, void* d_ws, size_t ws_size,
                              hipStream_t stream) {
  const float* x    = (const float*)d_in[0];
  const float* dwk  = (const float*)d_in[1];
  const float* pwk  = (const float*)d_in[2];
  const float* sepb = (const float*)d_in[3];
  const float* qkvk = (const float*)d_in[4];
  const float* qkvb = (const float*)d_in[5];
  const float* attk = (const float*)d_in[6];
  const float* attb = (const float*)d_in[7];
  const float* relw = (const float*)d_in[8];
  const float* relh = (const float*)d_in[9];
  float* out = (float*)d_out;

  float* ws_dw   = (float*)d_ws;                            // 8 MB
  float* ws_qkv  = ws_dw + (size_t)BATCH * HW * CIN;        // 12 MB
  float* ws_attn = ws_qkv + (size_t)BATCH * HW * 384;       // 4 MB

  const int M = BATCH * HW;

  dw_conv3x3<<<(BATCH * HW * CIN) / 256, 256, 0, stream>>>(x, dwk, ws_dw);
  gemm_bias_wmma<<<dim3(128 / 16, M / 128), 256, 0, stream>>>(
      ws_dw, CIN, pwk, 128, sepb, out, 256, 0, CIN);
  gemm_bias_wmma<<<dim3(384 / 16, M / 128), 256, 0, stream>>>(
      x, CIN, qkvk, 384, qkvb, ws_qkv, 384, 0, CIN);
  attn_flash_wmma<<<dim3(HW / 128, BATCH * NHEAD), 256, 0, stream>>>(
      ws_qkv, relw, relh, ws_attn);
  gemm_bias_wmma<<<dim3(128 / 16, M / 128), 256, 0, stream>>>(
      ws_attn, 128, attk, 128, attb, out, 256, 128, 128);
}